// LengthRegulator_67783173865529
// MI455X (gfx1250) — compile-verified
//
#include <hip/hip_runtime.h>
#include <hip/hip_bf16.h>
#include <stdint.h>

// LengthRegulator for MI455X (gfx1250).
// out[b,m,:] = inputs[b, token(m), :]  (pure row gather; zero FLOPs)
// Traffic ~63 MB -> ~2.7us floor at 23.3 TB/s. Data movement done on the
// CDNA5 async global<->LDS path (ASYNCcnt) instead of VGPR bounce.

#define B_ 16
#define T_ 512
#define D_ 384   // 384 floats = 1536 B per row = 96 lanes * 16 B

// ---------------------------------------------------------------------------
// Kernel 1: per-batch reps + inclusive scan (LDS), emit exclusive cumsum
// (T_+1 ints per batch) into workspace, and output_lens into d_out tail.
// ---------------------------------------------------------------------------
__global__ void lr_scan_kernel(const float* __restrict__ dur,
                               int* __restrict__ cum,
                               int* __restrict__ lens_out) {
  __shared__ int s[T_];
  const int b = blockIdx.x;
  const int t = threadIdx.x;

  // reps = floor(d + 0.5); durations are small non-negative integers in fp32.
  const int rep = (int)floorf(dur[b * T_ + t] + 0.5f);
  s[t] = rep;
  __syncthreads();

  // Hillis-Steele inclusive scan over 512 elements (9 steps).
  #pragma unroll
  for (int off = 1; off < T_; off <<= 1) {
    const int v = (t >= off) ? s[t - off] : 0;
    __syncthreads();
    s[t] += v;
    __syncthreads();
  }

  const int incl = s[t];
  cum[b * (T_ + 1) + t] = incl - rep;     // exclusive prefix sum
  if (t == T_ - 1) {
    cum[b * (T_ + 1) + T_] = incl;        // total = output_lens[b]
    if (lens_out) lens_out[b] = incl;     // int32 bits, matching reference dtype
  }
}

// ---------------------------------------------------------------------------
// Kernel 2: one block per output frame (m, b). Binary search the cumsum for
// the owning token, then DMA the 1536B row: global -> LDS -> global using the
// gfx1250 async LDS instructions (tracked by ASYNCcnt).
// ---------------------------------------------------------------------------
__global__ void __launch_bounds__(96)
lr_gather_kernel(const float* __restrict__ in,
                 const int* __restrict__ cum,
                 float* __restrict__ out,
                 int max_len) {
  __shared__ float row[D_];   // 1536 B staging buffer for the async DMA
  __shared__ int tok_s;

  const int m   = blockIdx.x;
  const int b   = blockIdx.y;
  const int tid = threadIdx.x;   // 0..95

  const int* __restrict__ c = cum + b * (T_ + 1);
  const int len = c[T_];

  float* dst = out + ((size_t)b * max_len + m) * D_;

  if (m >= len) {
    // Past this batch's real length: reference pads with zeros (d_out is
    // poisoned, so we must write them every call).
    float4 z = make_float4(0.f, 0.f, 0.f, 0.f);
    ((float4*)dst)[tid] = z;   // global_store_b128
    return;
  }

  // upper_bound over exclusive cumsum c[0..T_]: find t with c[t] <= m < c[t+1].
  if (tid == 0) {
    int lo = 0, hi = T_;                  // invariant: c[lo] <= m < c[hi]
    while (hi - lo > 1) {
      const int mid = (lo + hi) >> 1;
      if (c[mid] <= m) lo = mid; else hi = mid;
    }
    tok_s = lo;
  }
  __syncthreads();
  const int tok = tok_s;

  // LDS byte offset of the staging buffer: low 32 bits of the flat address of
  // a __shared__ object are the wave-relative LDS address (ISA 10.2 aperture).
  const unsigned lds_off = (unsigned)(uintptr_t)(&row[0]) + (unsigned)(tid * 16);

  const char* src = (const char*)(in + ((size_t)b * T_ + tok) * D_) + tid * 16;
  char*       dsc = (char*)dst + tid * 16;

  // Async DMA: global -> LDS (16 B/lane, 512 B/wave per instruction).
  asm volatile("global_load_async_to_lds_b128 %0, %1, off"
               :: "v"(lds_off), "v"(src) : "memory");
  // Async LDS accesses may complete out of order; drain before reading back.
  asm volatile("s_wait_asynccnt 0" ::: "memory");
  // Async DMA: LDS -> global.
  asm volatile("global_store_async_from_lds_b128 %0, %1, off"
               :: "v"(dsc), "v"(lds_off) : "memory");
  asm volatile("s_wait_asynccnt 0" ::: "memory");
}

// ---------------------------------------------------------------------------
// Host launcher.
// ---------------------------------------------------------------------------
extern "C" void kernel_launch(void* const* d_in, const int* in_sizes, int n_in,
                              void* d_out, int out_size, void* d_ws, size_t ws_size,
                              hipStream_t stream) {
  const float* inputs = (const float*)d_in[0];   // [B, T_IN, D] fp32
  const float* durs   = (const float*)d_in[1];   // [B, T_IN]    fp32
  float* out = (float*)d_out;

  // Recover data-dependent max_len from the harness-resolved output size.
  // Tuple output = [B, max_len, D] frames + [B] lengths, concatenated flat.
  int max_len;
  int* lens_out = nullptr;
  if (out_size % (B_ * D_) == 0) {
    max_len = out_size / (B_ * D_);                       // lens not in d_out
  } else {
    max_len = (out_size - B_) / (B_ * D_);
    lens_out = (int*)(out + (size_t)B_ * max_len * D_);   // int32 bits in tail
  }

  int* cum = (int*)d_ws;   // B * (T_+1) ints = ~33 KB of scratch

  lr_scan_kernel<<<B_, T_, 0, stream>>>(durs, cum, lens_out);

  dim3 grid(max_len, B_);
  lr_gather_kernel<<<grid, 96, 0, stream>>>(inputs, cum, out, max_len);
}